// _GraphResidualBlock_82076825026573
// MI455X (gfx1250) — compile-verified
//
#include <hip/hip_runtime.h>

#define N_NODES 10000
#define N_EDGES 640000
#define D_IN 128
#define D_OUT 256
#define BN_EPS 1e-5f

typedef __attribute__((ext_vector_type(2))) float v2f;
typedef __attribute__((ext_vector_type(8))) float v8f;

// ---------------- BatchNorm statistics (2-level reduction) ----------------
__global__ void bn_partial(const float* __restrict__ x,
                           float* __restrict__ bnsum,
                           float* __restrict__ bnsumsq) {
    const int col = threadIdx.x;            // 0..127
    const int r0 = blockIdx.x * 400;        // 25 blocks x 400 rows
    float s = 0.f, s2 = 0.f;
    for (int r = r0; r < r0 + 400; ++r) {
        float v = x[r * D_IN + col];
        s += v;
        s2 += v * v;
    }
    atomicAdd(&bnsum[col], s);
    atomicAdd(&bnsumsq[col], s2);
}

__global__ void bn_finalize(const float* __restrict__ bnsum,
                            const float* __restrict__ bnsumsq,
                            float* __restrict__ mean,
                            float* __restrict__ invstd) {
    const int c = threadIdx.x;
    const float m = bnsum[c] * (1.0f / (float)N_NODES);
    const float v = bnsumsq[c] * (1.0f / (float)N_NODES) - m * m;
    mean[c] = m;
    invstd[c] = rsqrtf(v + BN_EPS);
}

__global__ void bn_apply(const float* __restrict__ x,
                         const float* __restrict__ mean,
                         const float* __restrict__ invstd,
                         const float* __restrict__ gamma,
                         const float* __restrict__ beta,
                         float* __restrict__ xn) {
    const unsigned idx = blockIdx.x * blockDim.x + threadIdx.x;
    if (idx >= (unsigned)(N_NODES * D_IN)) return;
    const int c = idx & (D_IN - 1);
    xn[idx] = (x[idx] - mean[c]) * invstd[c] * gamma[c] + beta[c];
}

// ---------------- Edge degree + inverse ----------------
__global__ void edge_count(const int* __restrict__ tgt, float* __restrict__ cnt) {
    const unsigned e = blockIdx.x * blockDim.x + threadIdx.x;
    if (e >= (unsigned)N_EDGES) return;
    atomicAdd(&cnt[tgt[e]], 1.0f);
}

__global__ void inv_count(const float* __restrict__ cnt, float* __restrict__ icnt) {
    const unsigned i = blockIdx.x * blockDim.x + threadIdx.x;
    if (i >= (unsigned)N_NODES) return;
    icnt[i] = 1.0f / fmaxf(cnt[i], 1.0f);
}

// ---------------- Edge scatter: float4 per thread ----------------
// One b128 load + 4 f32 atomics per thread; edge index is wave-uniform
// (lanes of a wave cover consecutive channels of the same edge).
__global__ void edge_scatter4(const float* __restrict__ feat,
                              const int* __restrict__ src,
                              const int* __restrict__ tgt,
                              float* __restrict__ msum,
                              int log2tpe,          // threads per edge: 5 (d=128) or 6 (d=256)
                              unsigned total_threads) {
    const unsigned idx = blockIdx.x * blockDim.x + threadIdx.x;
    if (idx >= total_threads) return;
    const int log2d = log2tpe + 2;
    const unsigned e = idx >> log2tpe;
    const unsigned c = (idx & ((1u << log2tpe) - 1u)) << 2;
    const unsigned s = (unsigned)src[e];
    const unsigned t = (unsigned)tgt[e];
    const float4 v = *(const float4*)(feat + ((size_t)s << log2d) + c);
    float* dst = msum + ((size_t)t << log2d) + c;
    atomicAdd(dst + 0, v.x);
    atomicAdd(dst + 1, v.y);
    atomicAdd(dst + 2, v.z);
    atomicAdd(dst + 3, v.w);
}

// ---------------- WMMA fp32 GEMMs (V_WMMA_F32_16X16X4_F32) ----------------
// One 16x16 output tile per wave32; 10000 rows = 625 row-tiles exactly, so no
// partial waves and EXEC stays all-ones (WMMA requirement).
// f32 A(16x4): lanes 0-15 -> M=lane, K={k,k+1}; lanes 16-31 -> M=lane-16, K={k+2,k+3}.
// B(4x16): same K split; N = lane&15. C/D: 8 VGPRs, lanes 16-31 carry rows M+8.
// Independent accumulator per constituent GEMM -> independent WMMA chains.

__global__ void __launch_bounds__(256) sage1_gemm(
        const float* __restrict__ msum1, const float* __restrict__ icnt,
        const float* __restrict__ xn,
        const float* __restrict__ Wl1, const float* __restrict__ Wr1,
        const float* __restrict__ bl1, float* __restrict__ h) {
    const int lane = threadIdx.x & 31;
    const int wave = threadIdx.x >> 5;
    const int tile = blockIdx.x * 8 + wave;       // 1250 blocks * 8 waves = 10000 tiles
    const int tiles_n = D_OUT / 16;               // 16
    const int tm = tile / tiles_n;
    const int tn = tile - tm * tiles_n;
    const int m0 = tm * 16, n0 = tn * 16;
    const int l15 = lane & 15;
    const int half = (lane >> 4) << 1;            // 0 or 2
    const int m = m0 + l15;
    const int n = n0 + l15;
    const float ic = icnt[m];
    const float* __restrict__ Aagg  = msum1 + (size_t)m * D_IN;
    const float* __restrict__ Aself = xn + (size_t)m * D_IN;
    const float* __restrict__ Bl = Wl1 + (size_t)n * D_IN;   // B[k][n] = Wl1[n][k]
    const float* __restrict__ Br = Wr1 + (size_t)n * D_IN;

    v8f acc0 = {}, acc1 = {};
#pragma unroll 4
    for (int k = 0; k < D_IN; k += 4) {
        const int kk = k + half;
        v2f a0 = {Aagg[kk] * ic, Aagg[kk + 1] * ic};
        v2f b0 = {Bl[kk], Bl[kk + 1]};
        acc0 = __builtin_amdgcn_wmma_f32_16x16x4_f32(false, a0, false, b0,
                                                     (short)0, acc0, false, false);
        v2f a1 = {Aself[kk], Aself[kk + 1]};
        v2f b1 = {Br[kk], Br[kk + 1]};
        acc1 = __builtin_amdgcn_wmma_f32_16x16x4_f32(false, a1, false, b1,
                                                     (short)0, acc1, false, false);
    }
    const v8f acc = acc0 + acc1;
    const float bias = bl1[n];
    const int rbase = m0 + ((lane >> 4) << 3);    // +8 rows for upper half-wave
#pragma unroll
    for (int v = 0; v < 8; ++v) {
        float val = acc[v] + bias;
        h[(size_t)(rbase + v) * D_OUT + n] = fmaxf(val, 0.0f);   // fused ReLU
    }
}

__global__ void __launch_bounds__(256) sage2_gemm(
        const float* __restrict__ msum2, const float* __restrict__ icnt,
        const float* __restrict__ h, const float* __restrict__ x,
        const float* __restrict__ Wl2, const float* __restrict__ Wr2,
        const float* __restrict__ Ws,
        const float* __restrict__ bl2, const float* __restrict__ bs,
        float* __restrict__ out) {
    const int lane = threadIdx.x & 31;
    const int wave = threadIdx.x >> 5;
    const int tile = blockIdx.x * 8 + wave;
    const int tiles_n = D_OUT / 16;
    const int tm = tile / tiles_n;
    const int tn = tile - tm * tiles_n;
    const int m0 = tm * 16, n0 = tn * 16;
    const int l15 = lane & 15;
    const int half = (lane >> 4) << 1;
    const int m = m0 + l15;
    const int n = n0 + l15;
    const float ic = icnt[m];
    const float* __restrict__ Aagg  = msum2 + (size_t)m * D_OUT;
    const float* __restrict__ Aself = h + (size_t)m * D_OUT;
    const float* __restrict__ Ax    = x + (size_t)m * D_IN;
    const float* __restrict__ Bl  = Wl2 + (size_t)n * D_OUT;
    const float* __restrict__ Br  = Wr2 + (size_t)n * D_OUT;
    const float* __restrict__ Bsh = Ws + (size_t)n * D_IN;

    v8f acc0 = {}, acc1 = {}, acc2 = {};
#pragma unroll 4
    for (int k = 0; k < D_OUT; k += 4) {          // agg2@Wl2^T and h@Wr2^T interleaved
        const int kk = k + half;
        v2f a0 = {Aagg[kk] * ic, Aagg[kk + 1] * ic};
        v2f b0 = {Bl[kk], Bl[kk + 1]};
        acc0 = __builtin_amdgcn_wmma_f32_16x16x4_f32(false, a0, false, b0,
                                                     (short)0, acc0, false, false);
        v2f a1 = {Aself[kk], Aself[kk + 1]};
        v2f b1 = {Br[kk], Br[kk + 1]};
        acc1 = __builtin_amdgcn_wmma_f32_16x16x4_f32(false, a1, false, b1,
                                                     (short)0, acc1, false, false);
    }
#pragma unroll 4
    for (int k = 0; k < D_IN; k += 4) {           // x @ Ws^T (residual shortcut)
        const int kk = k + half;
        v2f a2 = {Ax[kk], Ax[kk + 1]};
        v2f b2 = {Bsh[kk], Bsh[kk + 1]};
        acc2 = __builtin_amdgcn_wmma_f32_16x16x4_f32(false, a2, false, b2,
                                                     (short)0, acc2, false, false);
    }
    const v8f acc = acc0 + acc1 + acc2;
    const float bias = bl2[n] + bs[n];
    const int rbase = m0 + ((lane >> 4) << 3);
#pragma unroll
    for (int v = 0; v < 8; ++v) {
        out[(size_t)(rbase + v) * D_OUT + n] = acc[v] + bias;
    }
}

// ---------------- Launch ----------------
extern "C" void kernel_launch(void* const* d_in, const int* in_sizes, int n_in,
                              void* d_out, int out_size, void* d_ws, size_t ws_size,
                              hipStream_t stream) {
    (void)in_sizes; (void)n_in; (void)out_size; (void)ws_size;
    const float* x     = (const float*)d_in[0];
    const int*   ei    = (const int*)d_in[1];
    const float* gamma = (const float*)d_in[2];
    const float* beta  = (const float*)d_in[3];
    const float* Wl1   = (const float*)d_in[4];
    const float* bl1   = (const float*)d_in[5];
    const float* Wr1   = (const float*)d_in[6];
    const float* Wl2   = (const float*)d_in[7];
    const float* bl2   = (const float*)d_in[8];
    const float* Wr2   = (const float*)d_in[9];
    const float* Ws    = (const float*)d_in[10];
    const float* bs    = (const float*)d_in[11];
    const int* src = ei;               // edge_index[0]
    const int* tgt = ei + N_EDGES;     // edge_index[1]

    // Workspace layout (floats). Front region is zero-initialized each call.
    // All large buffers are 16-float aligned -> b128 loads legal.
    float* ws      = (float*)d_ws;
    float* bnsum   = ws + 0;           //    128
    float* bnsumsq = ws + 128;         //    128
    float* cnt     = ws + 256;         //  10000
    float* msum1   = ws + 10256;       // 1,280,000
    float* msum2   = ws + 1290256;     // 2,560,000
    const size_t zero_floats = 3850256;
    float* mean    = ws + 3850256;     //    128
    float* invstd  = ws + 3850384;     //    128
    float* icnt    = ws + 3850512;     //  10000
    float* xn      = ws + 3860512;     // 1,280,000
    float* h       = ws + 5140512;     // 2,560,000
    float* out     = (float*)d_out;

    hipMemsetAsync(d_ws, 0, zero_floats * sizeof(float), stream);

    // BatchNorm (training-mode batch stats, biased variance)
    bn_partial<<<25, 128, 0, stream>>>(x, bnsum, bnsumsq);
    bn_finalize<<<1, 128, 0, stream>>>(bnsum, bnsumsq, mean, invstd);
    bn_apply<<<(N_NODES * D_IN + 255) / 256, 256, 0, stream>>>(x, mean, invstd, gamma, beta, xn);

    // Edge degrees (shared by both convs)
    edge_count<<<(N_EDGES + 255) / 256, 256, 0, stream>>>(tgt, cnt);
    inv_count<<<(N_NODES + 255) / 256, 256, 0, stream>>>(cnt, icnt);

    // SAGEConv1: scatter xn -> msum1 (float4 granularity), then fused
    // (agg@Wl1^T + xn@Wr1^T + bl1, ReLU)
    edge_scatter4<<<(N_EDGES * 32) / 256, 256, 0, stream>>>(
        xn, src, tgt, msum1, 5, (unsigned)(N_EDGES * 32));
    sage1_gemm<<<1250, 256, 0, stream>>>(msum1, icnt, xn, Wl1, Wr1, bl1, h);

    // SAGEConv2 + residual: scatter h -> msum2, then fused
    // (agg@Wl2^T + h@Wr2^T + x@Ws^T + bl2 + bs)
    edge_scatter4<<<(N_EDGES * 64) / 256, 256, 0, stream>>>(
        h, src, tgt, msum2, 6, (unsigned)(N_EDGES * 64));
    sage2_gemm<<<1250, 256, 0, stream>>>(msum2, icnt, h, x, Wl2, Wr2, Ws, bl2, bs, out);
}